// IntegrateSPM_30305289240807
// MI455X (gfx1250) — compile-verified
//
#include <hip/hip_runtime.h>

typedef __attribute__((ext_vector_type(2))) float v2f;
typedef __attribute__((ext_vector_type(8))) float v8f;

#define NCH   19          // states per chain
#define NST   38          // total state components
#define BATCH 4096
#define KSEQ  1024
#define SBLK  16          // timesteps per GEMM block
#define WCOLS 608         // NST * SBLK
#define KPAD  64          // padded K (we only iterate 14 chunks = 56)

__device__ __forceinline__ int nn_idx(int i, int j) { return i * NCH + j; }

// ---------------------------------------------------------------------------
// Precompute kernel (1 block): build per-step affine map M = I + hA + h^2A^2/2
// + h^3A^3/6 + h^4A^4/24 and q = h(I + hA/2 + h^2A^2/6 + h^3A^3/24) b for both
// chains, then fill the time-block mixing matrix W (KPAD x WCOLS):
//   col = c*16 + jj  (state component c, time offset jj)
//   W[k][col]      = M^{jj+1}[c,k]          for k in [0,38)   (block-diagonal)
//   W[38+s][col]   = (M^{jj-s} q)[c]        for s <= jj
//   other rows = 0
// ---------------------------------------------------------------------------
__global__ __launch_bounds__(384)
void spm_precompute(const float* __restrict__ An_g, const float* __restrict__ Bn_g,
                    const float* __restrict__ Ap_g, const float* __restrict__ Bp_g,
                    float* __restrict__ W)
{
    const float Hc = 0.01f;
    __shared__ float sA[NCH*NCH], sA2[NCH*NCH], sA3[NCH*NCH], sA4[NCH*NCH];
    __shared__ float sM[2][NCH*NCH];   // per-step matrix, both chains
    __shared__ float sP[2][NCH*NCH];   // current power M^j
    __shared__ float sPt[2][NCH*NCH];  // temp
    __shared__ float sq[2][NCH], sr[2][NCH], srt[2][NCH];

    const int t = threadIdx.x;

    // zero W entirely (covers pad rows and the s>jj triangle)
    for (int i = t; i < KPAD * WCOLS; i += 384) W[i] = 0.0f;

    for (int chain = 0; chain < 2; ++chain) {
        const float* Ag = chain ? Ap_g : An_g;
        const float* Bg = chain ? Bp_g : Bn_g;
        if (t < NCH*NCH) sA[t] = Ag[t];
        __syncthreads();
        if (t < NCH*NCH) { int i = t / NCH, j = t % NCH; float s = 0.f;
            for (int k = 0; k < NCH; ++k) s += sA[nn_idx(i,k)] * sA[nn_idx(k,j)];
            sA2[t] = s; }
        __syncthreads();
        if (t < NCH*NCH) { int i = t / NCH, j = t % NCH; float s = 0.f;
            for (int k = 0; k < NCH; ++k) s += sA2[nn_idx(i,k)] * sA[nn_idx(k,j)];
            sA3[t] = s; }
        __syncthreads();
        if (t < NCH*NCH) { int i = t / NCH, j = t % NCH; float s = 0.f;
            for (int k = 0; k < NCH; ++k) s += sA3[nn_idx(i,k)] * sA[nn_idx(k,j)];
            sA4[t] = s; }
        __syncthreads();
        if (t < NCH*NCH) { int i = t / NCH, j = t % NCH;
            sM[chain][t] = ((i == j) ? 1.0f : 0.0f)
                         + Hc * sA[t]
                         + (Hc*Hc * 0.5f) * sA2[t]
                         + (Hc*Hc*Hc / 6.0f) * sA3[t]
                         + (Hc*Hc*Hc*Hc / 24.0f) * sA4[t]; }
        if (t < NCH) {
            float ab = 0.f, a2b = 0.f, a3b = 0.f;
            for (int k = 0; k < NCH; ++k) {
                float bk = Bg[k];
                ab  += sA [nn_idx(t,k)] * bk;
                a2b += sA2[nn_idx(t,k)] * bk;
                a3b += sA3[nn_idx(t,k)] * bk;
            }
            sq[chain][t] = Hc * (Bg[t] + (Hc*0.5f)*ab + (Hc*Hc/6.0f)*a2b
                                 + (Hc*Hc*Hc/24.0f)*a3b);
        }
        __syncthreads();
    }

    // ---- drive-coefficient part: r_m = M^m q ----
    if (t < NCH) { sr[0][t] = sq[0][t]; sr[1][t] = sq[1][t]; }
    __syncthreads();
    for (int m = 0; m < SBLK; ++m) {
        for (int idx = t; idx < NST * SBLK; idx += 384) {
            int c = idx / SBLK, s = idx % SBLK;
            if (s <= (SBLK - 1) - m) {
                float val = (c < NCH) ? sr[0][c] : sr[1][c - NCH];
                W[(NST + s) * WCOLS + c * SBLK + (m + s)] = val;
            }
        }
        __syncthreads();
        if (t < NCH) {
            float s0 = 0.f, s1 = 0.f;
            for (int k = 0; k < NCH; ++k) {
                s0 += sM[0][nn_idx(t,k)] * sr[0][k];
                s1 += sM[1][nn_idx(t,k)] * sr[1][k];
            }
            srt[0][t] = s0; srt[1][t] = s1;
        }
        __syncthreads();
        if (t < NCH) { sr[0][t] = srt[0][t]; sr[1][t] = srt[1][t]; }
        __syncthreads();
    }

    // ---- state-propagation part: M^j for j = 1..16 ----
    if (t < NCH*NCH) { sP[0][t] = sM[0][t]; sP[1][t] = sM[1][t]; }
    __syncthreads();
    for (int j = 1; j <= SBLK; ++j) {
        for (int idx = t; idx < NST * NST; idx += 384) {
            int c = idx / NST, k = idx % NST;
            float val = 0.f;
            if (c < NCH && k < NCH)          val = sP[0][nn_idx(c, k)];
            else if (c >= NCH && k >= NCH)   val = sP[1][nn_idx(c - NCH, k - NCH)];
            W[k * WCOLS + c * SBLK + (j - 1)] = val;
        }
        __syncthreads();
        if (j < SBLK) {
            if (t < NCH*NCH) { int i = t / NCH, jj = t % NCH;
                float s0 = 0.f, s1 = 0.f;
                for (int k = 0; k < NCH; ++k) {
                    s0 += sP[0][nn_idx(i,k)] * sM[0][nn_idx(k,jj)];
                    s1 += sP[1][nn_idx(i,k)] * sM[1][nn_idx(k,jj)];
                }
                sPt[0][t] = s0; sPt[1][t] = s1; }
            __syncthreads();
            if (t < NCH*NCH) { sP[0][t] = sPt[0][t]; sP[1][t] = sPt[1][t]; }
            __syncthreads();
        }
    }
}

// ---------------------------------------------------------------------------
// Per-time-block GEMM: out[b, :, T..T+15] = [x_T | i_blk | 0] @ W, via
// V_WMMA_F32_16X16X4_F32. One wave = 16 batch rows x one state component
// (16 time columns). Carry x_{T+16} (jj == 15 column) goes to stateNext.
// ---------------------------------------------------------------------------
__global__ __launch_bounds__(256)
void spm_block_gemm(const float* __restrict__ stateCur,
                    float* __restrict__ stateNext,
                    const float* __restrict__ W,
                    const float* __restrict__ iseq,
                    float* __restrict__ out,
                    int T)
{
    const int wave  = threadIdx.x >> 5;            // 0..7
    const int lane  = threadIdx.x & 31;
    const int ct    = blockIdx.x;                  // state component 0..37
    const int b0    = (blockIdx.y * 8 + wave) * 16; // batch tile base
    const int row   = lane & 15;                   // A-frag: batch row in tile
    const int khalf = lane >> 4;                   // A/B frag: K half select

    const int b = b0 + row;
    v8f acc = {0.f, 0.f, 0.f, 0.f, 0.f, 0.f, 0.f, 0.f};

#pragma unroll
    for (int chunk = 0; chunk < 14; ++chunk) {     // K = 56
        const int k = chunk * 4 + khalf * 2;       // this lane's K pair
        v2f a;
        if (k < NST) {                             // state part (38 even: pair safe)
            a = *(const v2f*)(stateCur + b * NST + k);
        } else if (k < NST + SBLK) {               // drive part
            a = *(const v2f*)(iseq + b * KSEQ + T + (k - NST));
        } else {                                   // zero pad (rows 54..55)
            a.x = 0.f; a.y = 0.f;
        }
        v2f bf;
        bf.x = W[(k + 0) * WCOLS + ct * SBLK + row];
        bf.y = W[(k + 1) * WCOLS + ct * SBLK + row];
        acc = __builtin_amdgcn_wmma_f32_16x16x4_f32(
                  false, a, false, bf, (short)0, acc, false, false);
    }

    // C/D layout: VGPR r -> M = r + 8*khalf, N = lane&15 (= time offset jj)
    const int n = lane & 15;
#pragma unroll
    for (int r = 0; r < 8; ++r) {
        const int m  = r + 8 * khalf;
        const int bb = b0 + m;
        out[(size_t)bb * (NST * KSEQ) + (size_t)ct * KSEQ + T + n] = acc[r];
        if (n == SBLK - 1)                          // carry x_{T+16}
            stateNext[bb * NST + ct] = acc[r];
    }
}

// ---------------------------------------------------------------------------
extern "C" void kernel_launch(void* const* d_in, const int* in_sizes, int n_in,
                              void* d_out, int out_size, void* d_ws, size_t ws_size,
                              hipStream_t stream)
{
    const float* x    = (const float*)d_in[0];   // (4096, 38)
    const float* iseq = (const float*)d_in[1];   // (4096, 1024)
    const float* An   = (const float*)d_in[2];   // (19, 19)
    const float* Bn   = (const float*)d_in[3];   // (19,)
    const float* Ap   = (const float*)d_in[4];   // (19, 19)
    const float* Bp   = (const float*)d_in[5];   // (19,)
    float* out = (float*)d_out;                  // (4096, 38, 1024)

    float* ws     = (float*)d_ws;
    float* stateA = ws;                          // 4096*38
    float* stateB = ws + BATCH * NST;            // 4096*38
    float* W      = ws + 2 * BATCH * NST;        // 64*608

    // x0 -> state buffer (identical layout)
    hipMemcpyAsync(stateA, x, (size_t)BATCH * NST * sizeof(float),
                   hipMemcpyDeviceToDevice, stream);

    spm_precompute<<<1, 384, 0, stream>>>(An, Bn, Ap, Bp, W);

    for (int blk = 0; blk < KSEQ / SBLK; ++blk) {
        const float* cur = (blk & 1) ? stateB : stateA;
        float*       nxt = (blk & 1) ? stateA : stateB;
        spm_block_gemm<<<dim3(NST, BATCH / (16 * 8)), 256, 0, stream>>>(
            cur, nxt, W, iseq, out, blk * SBLK);
    }
}